// ProbsNet_35287451304896
// MI455X (gfx1250) — compile-verified
//
#include <hip/hip_runtime.h>

typedef __attribute__((ext_vector_type(2))) float v2f;
typedef __attribute__((ext_vector_type(8))) float v8f;

#define MDIM 500000
#define CHUNKS (MDIM / 4)          // 4 m-rows per WMMA step
#define BPC 128                    // blocks per (group) combo
#define WAVES_PER_BLOCK 8          // 256 threads, wave32
#define WAVES_PER_COMBO (BPC * WAVES_PER_BLOCK)

// Streams sigmoid(B*(base+ST[g,m])) against W[g,m,0..20], accumulating the
// 21-wide matvec in the WMMA f32 16x16x4 accumulator.
//   A[r,k]   = s[m0+k]  (replicated over the 16 rows)
//   B[k,n]   = W[(m0+k)*21 + pbase + n]
//   D[r,n]  += sum_k s[m0+k] * W[m0+k, pbase+n]   (all rows identical)
// Layout (ISA 7.12.2, symmetric A/B pattern): K lives in VGPR (k&1), half (k>>1).
__global__ __launch_bounds__(256) void probs_stream_wmma(
    const float* __restrict__ ST, const float* __restrict__ W,
    const float* __restrict__ BEV, const float* __restrict__ BEV_p,
    const float* __restrict__ Bp, float* __restrict__ tmp /*84 floats*/)
{
    const int g            = blockIdx.x & 3;
    const int blockInCombo = blockIdx.x >> 2;
    const int wave         = threadIdx.x >> 5;
    const int lane         = threadIdx.x & 31;
    const int h            = lane >> 4;   // half: selects K = 2h, 2h+1
    const int n            = lane & 15;   // column within p-tile

    // Wave-uniform by construction; force into an SGPR so the chunk loop is
    // pure SALU control flow (no per-lane exec masking around the WMMAs).
    const int waveId = __builtin_amdgcn_readfirstlane(
        blockInCombo * WAVES_PER_BLOCK + wave);

    const float Bv   = Bp[0];
    const float base = fmaxf(BEV_p[0], 0.0f) * BEV[0];

    const float* STg = ST + (size_t)g * MDIM;
    const float* Wg  = W  + (size_t)g * MDIM * 21;

    v8f c0 = {};   // p = 0..15
    v8f c1 = {};   // p = 16..20 (cols 5..15 are discarded slack)

#pragma unroll 2
    for (int j = waveId; j < CHUNKS; j += WAVES_PER_COMBO) {
        const int r = 4 * j + 2 * h;                 // row index for this half
        v2f st = *(const v2f*)(STg + r);             // broadcast b64 per half
        // sigmoid via v_exp_f32 + v_rcp_f32 (no IEEE divide in the hot loop)
        float s0 = __builtin_amdgcn_rcpf(1.0f + __expf(-Bv * (base + st.x)));
        float s1 = __builtin_amdgcn_rcpf(1.0f + __expf(-Bv * (base + st.y)));
        v2f a; a.x = s0; a.y = s1;                   // A: K=2h, K=2h+1

        const float* wr = Wg + (size_t)r * 21 + n;   // row r, col n (NT stream)
        v2f b1; b1.x = __builtin_nontemporal_load(wr);
                b1.y = __builtin_nontemporal_load(wr + 21);
        v2f b2; b2.x = __builtin_nontemporal_load(wr + 16);
                b2.y = __builtin_nontemporal_load(wr + 37);

        c0 = __builtin_amdgcn_wmma_f32_16x16x4_f32(false, a, false, b1,
                                                   (short)0, c0, false, false);
        c1 = __builtin_amdgcn_wmma_f32_16x16x4_f32(false, a, false, b2,
                                                   (short)0, c1, false, false);
    }

    // D row 0: VGPR0 of C/D holds (M=0, N=lane) in lanes 0..15.
    float v0 = c0[0];
    float v1 = c1[0];
    if (lane < 16) atomicAdd(tmp + g * 21 + lane,      v0);
    if (lane < 5)  atomicAdd(tmp + g * 21 + 16 + lane, v1);
}

__global__ void probs_init(float* __restrict__ tmp)
{
    int i = threadIdx.x;
    if (i < 168) tmp[i] = 0.0f;   // tmp[2][84]
}

// Reproduces calc_probs ordering: per i: [p_i, for j: [p_i*p_j, for k: p_i*p_j*p_k]]
__global__ void probs_finalize(const float* __restrict__ pp,
                               const float* __restrict__ tmp,
                               float* __restrict__ out)
{
    if (threadIdx.x != 0) return;
    float acc_sum = 0.0f;
    for (int r = 0; r < 5; ++r) {
        float x[4], p[4];
        float mx = -1e30f;
        for (int i = 0; i < 4; ++i) { x[i] = pp[r * 4 + i]; mx = fmaxf(mx, x[i]); }
        float s = 0.0f;
        for (int i = 0; i < 4; ++i) { p[i] = __expf(x[i] - mx); s += p[i]; }
        float inv = 1.0f / s;
        for (int i = 0; i < 4; ++i) p[i] *= inv;

        const float* t = (r == 0) ? tmp : (tmp + 84);
        float acc = 0.0f;
        for (int i = 0; i < 4; ++i) {
            const float* ti = t + i * 21;
            acc += p[i] * ti[0];
            int idx = 1;
            for (int jj = 0; jj < 4; ++jj) {
                float pij = p[i] * p[jj];
                acc += pij * ti[idx++];
                for (int k = 0; k < 4; ++k) acc += pij * p[k] * ti[idx++];
            }
        }
        acc_sum += acc;
    }
    out[0] = acc_sum * 0.2f;   // mean of 5
}

extern "C" void kernel_launch(void* const* d_in, const int* in_sizes, int n_in,
                              void* d_out, int out_size, void* d_ws, size_t ws_size,
                              hipStream_t stream)
{
    (void)in_sizes; (void)n_in; (void)out_size; (void)ws_size;
    const float* BEV  = (const float*)d_in[0];
    const float* ST0  = (const float*)d_in[1];
    const float* W0   = (const float*)d_in[2];
    const float* ST1  = (const float*)d_in[3];
    const float* W1   = (const float*)d_in[4];
    const float* PP   = (const float*)d_in[5];
    const float* BEVp = (const float*)d_in[6];
    const float* B    = (const float*)d_in[7];

    float* tmp = (float*)d_ws;        // [2][84] partial sums
    float* out = (float*)d_out;       // 1 float

    probs_init<<<dim3(1), dim3(256), 0, stream>>>(tmp);
    probs_stream_wmma<<<dim3(4 * BPC), dim3(256), 0, stream>>>(ST0, W0, BEV, BEVp, B, tmp);
    probs_stream_wmma<<<dim3(4 * BPC), dim3(256), 0, stream>>>(ST1, W1, BEV, BEVp, B, tmp + 84);
    probs_finalize<<<dim3(1), dim3(32), 0, stream>>>(PP, tmp, out);
}